// ModelNew_29240137351227
// MI455X (gfx1250) — compile-verified
//
#include <hip/hip_runtime.h>

// Rebased linear attention (squared scaled QK, causal, row-sum normalized)
// for MI455X / gfx1250. All matmul work on V_WMMA_F32_16X16X4_F32 (full fp32
// precision; the reference squares scores, so low-precision A/B is unsafe).

typedef __attribute__((ext_vector_type(2))) float v2f;
typedef __attribute__((ext_vector_type(8))) float v8f;

#define EPSV 1e-6f

__global__ __launch_bounds__(256) void rebased_attn_kernel(
    const float* __restrict__ Q, const float* __restrict__ K,
    const float* __restrict__ V, float* __restrict__ O,
    int H, int S)
{
  constexpr int D  = 64;   // head dim (fixed by reference)
  constexpr int KP = 68;   // padded LDS row stride for K/V tiles (dwords) -> 0 bank conflicts
  constexpr int SP = 20;   // padded LDS row stride for S2 scratch (dwords) -> 0 bank conflicts
  constexpr int NW = 8;    // waves per block
  constexpr int BR = 16 * NW;  // 128 query rows per block

  __shared__ __align__(16) float lk[16 * KP];
  __shared__ __align__(16) float lv[16 * KP];
  __shared__ __align__(16) float ls[NW * 16 * SP];

  const int NQB  = S / BR;
  const int head = blockIdx.x % H;
  const int qblk = (NQB - 1) - (blockIdx.x / H);  // heaviest blocks first

  const int tid  = threadIdx.x;
  const int wave = tid >> 5;
  const int lane = tid & 31;
  const int half = lane >> 4;   // K-half / M-half selector (wave32 layouts)
  const int ln   = lane & 15;   // M (for A), N (for B/C/D)

  const float* qh = Q + (size_t)head * S * D;
  const float* kh = K + (size_t)head * S * D;
  const float* vh = V + (size_t)head * S * D;
  float*       oh = O + (size_t)head * S * D;

  const int qrow0 = qblk * BR + wave * 16;

  // ---- Q tile -> A-fragments for f32 16x16x4 WMMA ----
  // chunk c covers d = 4c..4c+3; this lane holds d0 = 4c + 2*half and d0+1.
  v2f qa[16];
  {
    const float* qrow = qh + (size_t)(qrow0 + ln) * D + 2 * half;
#pragma unroll
    for (int c = 0; c < 16; ++c)
      qa[c] = *(const v2f*)(qrow + 4 * c);
  }

  v8f  oacc[4];
  float zacc[8];
#pragma unroll
  for (int t = 0; t < 4; ++t) oacc[t] = (v8f){};
#pragma unroll
  for (int r = 0; r < 8; ++r) zacc[r] = 0.f;

  // wave-uniform causal extents (readfirstlane -> scalar branch, EXEC stays all-1s for WMMA)
  const int my_jt  = __builtin_amdgcn_readfirstlane(qblk * NW + wave);
  const int blk_jt = qblk * NW + (NW - 1);

  // cooperative K/V staging: one float4 of K and one of V per thread
  const int crow = tid >> 4;         // 0..15
  const int ccol = (tid & 15) << 2;  // 0,4,...,60

  float4 kreg = *(const float4*)(kh + (size_t)crow * D + ccol);  // tile j=0
  float4 vreg = *(const float4*)(vh + (size_t)crow * D + ccol);

  float* s2b = ls + wave * 16 * SP;        // per-wave private scratch
  const float scale2 = 1.0f / (float)D;    // (D^-1/2)^2 folded into the square

  for (int j = 0; j <= blk_jt; ++j) {
    __syncthreads();                       // prior iteration's LDS reads done
    *(float4*)(lk + crow * KP + ccol) = kreg;
    *(float4*)(lv + crow * KP + ccol) = vreg;
    if (j < blk_jt) {                      // software pipeline: fetch tile j+1 now
      const size_t off = (size_t)((j + 1) * 16 + crow) * D + ccol;
      kreg = *(const float4*)(kh + off);
      vreg = *(const float4*)(vh + off);
    }
    __syncthreads();                       // K/V tile visible to all 8 waves

    if (j <= my_jt) {
      // ---- S = Q * K_j^T : 16 chained f32 WMMAs over D=64 ----
      v8f sacc = (v8f){};
#pragma unroll
      for (int c = 0; c < 16; ++c) {
        // B[d][n] = K[n][d]; lane holds rows d0 = 4c+2*half, d0+1 (contiguous -> ds b64)
        v2f kb = *(const v2f*)(lk + ln * KP + 4 * c + 2 * half);
        sacc = __builtin_amdgcn_wmma_f32_16x16x4_f32(
            false, qa[c], false, kb, (short)0, sacc, false, false);
      }

      // ---- square + scale + causal mask; accumulate z; spill S2 to LDS ----
      const bool diag = (j == my_jt);
#pragma unroll
      for (int r = 0; r < 8; ++r) {
        const int m = r + 8 * half;        // C-layout row
        float s  = sacc[r];
        float s2 = s * s * scale2;
        if (diag && ln > m) s2 = 0.f;      // key index > query index on diagonal tile
        zacc[r] += s2;
        s2b[m * SP + ln] = s2;             // C-layout -> row-major scratch
      }
      __builtin_amdgcn_wave_barrier();     // keep DS store->load order (same wave, in-order DS)

      // ---- O += S2 * V_j : 4 K-chunks x 4 N-tiles of f32 WMMA ----
#pragma unroll
      for (int kc = 0; kc < 4; ++kc) {
        const int k0 = 4 * kc + 2 * half;
        v2f a = *(const v2f*)(s2b + ln * SP + k0);   // A-layout gather of S2
#pragma unroll
        for (int t = 0; t < 4; ++t) {
          v2f b;                                      // B[k][n] = V[k][16t+n]
          b[0] = lv[k0 * KP + 16 * t + ln];
          b[1] = lv[(k0 + 1) * KP + 16 * t + ln];
          oacc[t] = __builtin_amdgcn_wmma_f32_16x16x4_f32(
              false, a, false, b, (short)0, oacc[t], false, false);
        }
      }
      __builtin_amdgcn_wave_barrier();
    }
  }

  // ---- finish z: butterfly over the 16 lanes sharing each row (wave32) ----
#pragma unroll
  for (int r = 0; r < 8; ++r) {
    float z = zacc[r];
    z += __shfl_xor(z, 1, 32);
    z += __shfl_xor(z, 2, 32);
    z += __shfl_xor(z, 4, 32);
    z += __shfl_xor(z, 8, 32);
    zacc[r] = 1.0f / (z + EPSV);
  }

  // ---- write O (coalesced 64B per 16-lane group) ----
#pragma unroll
  for (int r = 0; r < 8; ++r) {
    const int row = qrow0 + r + 8 * half;
    float* orow = oh + (size_t)row * D;
#pragma unroll
    for (int t = 0; t < 4; ++t)
      orow[16 * t + ln] = oacc[t][r] * zacc[r];
  }
}

extern "C" void kernel_launch(void* const* d_in, const int* in_sizes, int n_in,
                              void* d_out, int out_size, void* d_ws, size_t ws_size,
                              hipStream_t stream) {
  (void)n_in; (void)d_ws; (void)ws_size; (void)in_sizes; (void)out_size;
  const float* q = (const float*)d_in[0];
  const float* k = (const float*)d_in[1];
  const float* v = (const float*)d_in[2];
  float* out = (float*)d_out;

  const int H = 16, S = 2048;              // B=1, D=64 fixed by reference
  const int blocks = H * (S / 128);        // 256 blocks x 256 threads (8 wave32)
  rebased_attn_kernel<<<blocks, 256, 0, stream>>>(q, k, v, out, H, S);
}